// PredictRetinaface_84293028151793
// MI455X (gfx1250) — compile-verified
//
#include <hip/hip_runtime.h>
#include <cstdint>

#define B_IMG 16
#define N_ANCH 67200
#define TOPK 1000
#define KPAD 1024
#define CAND 4096
#define NBINS 4096
#define F_THR_CONF 0.5f
#define F_THR_NMS 0.4f
#define F_VAR0 0.1f
#define F_VAR1 0.2f
#define F_CLS_OFF 4.0f

typedef float v2f __attribute__((ext_vector_type(2)));
typedef float v8f __attribute__((ext_vector_type(8)));
typedef unsigned int v4u __attribute__((ext_vector_type(4)));
typedef int v4i __attribute__((ext_vector_type(4)));
typedef int v8i __attribute__((ext_vector_type(8)));

// ---------------------------------------------------------------- zero sup
__global__ void k_zero(unsigned* __restrict__ p, int n) {
    int i = blockIdx.x * blockDim.x + threadIdx.x;
    if (i < n) p[i] = 0u;
}

// ------------------------------------------------- pass 1: masked scores only
__global__ void k_scores(const float* __restrict__ p_conf, float* __restrict__ msc) {
    int i = blockIdx.x * blockDim.x + threadIdx.x;
    if (i < B_IMG * N_ANCH) {
        float x = p_conf[(size_t)i * 3];
        float s = 1.f / (1.f + __expf(-x));
        msc[i] = (x >= 0.f) ? s : 0.f;   // sigmoid(x) >= 0.5  <=>  x >= 0
    }
}

// -------------------------- pass 2: per-image histogram select + bitonic sort
__global__ void __launch_bounds__(1024)
k_topk(const float* __restrict__ msc, int* __restrict__ topIdx,
       float* __restrict__ topScore) {
    __shared__ unsigned hist[NBINS];
    __shared__ unsigned long long keys[CAND];
    __shared__ unsigned s_cnt;
    __shared__ int s_cut;
    const int b = blockIdx.x;
    const int tid = threadIdx.x;
    const float* sc = msc + (size_t)b * N_ANCH;

    for (int i = tid; i < NBINS; i += blockDim.x) hist[i] = 0u;
    if (tid == 0) { s_cnt = 0u; s_cut = 0; }
    __syncthreads();

    for (int i = tid; i < N_ANCH; i += blockDim.x) {
        float s = sc[i];
        if (s > 0.f) {
            int bin = (int)(s * (float)NBINS);
            bin = bin > (NBINS - 1) ? (NBINS - 1) : bin;
            atomicAdd(&hist[bin], 1u);
        }
    }
    __syncthreads();

    if (tid == 0) {
        unsigned cum = 0; int cb = 0;
        for (int bn = NBINS - 1; bn >= 0; --bn) {
            cum += hist[bn];
            if (cum >= TOPK) { cb = bn; break; }
        }
        s_cut = cb;
    }
    __syncthreads();
    const int cut = s_cut;

    for (int i = tid; i < N_ANCH; i += blockDim.x) {
        float s = sc[i];
        if (s > 0.f && (int)(s * (float)NBINS) >= cut) {
            unsigned pos = atomicAdd(&s_cnt, 1u);
            if (pos < CAND)
                keys[pos] = ((unsigned long long)__float_as_uint(s) << 32) |
                            (unsigned long long)(0xFFFFFFFFu - (unsigned)i);
        }
    }
    __syncthreads();
    unsigned cnt = s_cnt < (unsigned)CAND ? s_cnt : (unsigned)CAND;
    for (unsigned i = cnt + tid; i < CAND; i += blockDim.x) keys[i] = 0ull;

    // bitonic sort, descending (score major, index-ascending tie-break)
    for (unsigned k = 2; k <= CAND; k <<= 1) {
        for (unsigned j = k >> 1; j > 0; j >>= 1) {
            __syncthreads();
            for (unsigned i = tid; i < CAND; i += blockDim.x) {
                unsigned ixj = i ^ j;
                if (ixj > i) {
                    unsigned long long a = keys[i], c = keys[ixj];
                    bool desc = ((i & k) == 0);
                    if (desc ? (a < c) : (a > c)) { keys[i] = c; keys[ixj] = a; }
                }
            }
        }
    }
    __syncthreads();
    for (int r = tid; r < KPAD; r += blockDim.x) {
        unsigned long long kk = keys[r];
        float s = __uint_as_float((unsigned)(kk >> 32));
        unsigned idx = 0xFFFFFFFFu - (unsigned)(kk & 0xFFFFFFFFull);
        if (kk == 0ull) { s = 0.f; idx = 0u; }
        topScore[b * KPAD + r] = s;
        topIdx[b * KPAD + r]   = (int)idx;
    }
}

// ---------------- pass 3: decode only the winners; write 4 outputs + boff ws
__global__ void __launch_bounds__(1024)
k_gather(const float* __restrict__ p_loc, const float* __restrict__ p_conf,
         const float* __restrict__ p_landms, const float* __restrict__ anchors,
         const int* __restrict__ topIdx, const float* __restrict__ topScore,
         float* __restrict__ out, float* __restrict__ boffw) {
    const int b = blockIdx.x;
    const int r = threadIdx.x;              // 0..1023
    const int idx = topIdx[b * KPAD + r];
    const float sc = topScore[b * KPAD + r];

    const float ax = anchors[idx * 4 + 0], ay = anchors[idx * 4 + 1];
    const float aw = anchors[idx * 4 + 2], ah = anchors[idx * 4 + 3];
    const size_t base = (size_t)b * N_ANCH + (size_t)idx;
    const float* lc = p_loc + base * 4;
    float cx = ax + lc[0] * F_VAR0 * aw;
    float cy = ay + lc[1] * F_VAR0 * ah;
    float w  = aw * __expf(lc[2] * F_VAR1);
    float h  = ah * __expf(lc[3] * F_VAR1);
    float x1 = cx - 0.5f * w, y1 = cy - 0.5f * h;
    float x2 = cx + 0.5f * w, y2 = cy + 0.5f * h;
    float c1 = p_conf[base * 3 + 1], c2 = p_conf[base * 3 + 2];
    float lab = (c1 >= c2) ? 1.f : 2.f;     // argmax(softmax)=argmax(logits), +1

    if (r < TOPK) {
        float* ob = out + (size_t)B_IMG * TOPK + ((size_t)b * TOPK + r) * 4;
        ob[0] = x1; ob[1] = y1; ob[2] = x2; ob[3] = y2;
        float* ok = out + (size_t)B_IMG * TOPK * 5 + ((size_t)b * TOPK + r) * 10;
        const float* lm = p_landms + base * 10;
#pragma unroll
        for (int k = 0; k < 5; ++k) {
            ok[2 * k + 0] = ax + lm[2 * k + 0] * F_VAR0 * aw;
            ok[2 * k + 1] = ay + lm[2 * k + 1] * F_VAR0 * ah;
        }
        out[(size_t)B_IMG * TOPK * 15 + b * TOPK + r] = lab;
        out[(size_t)B_IMG * TOPK * 16 + b * TOPK + r] = sc;
    }
    float off = lab * F_CLS_OFF;
    float* bo = boffw + ((size_t)b * KPAD + r) * 4;
    bo[0] = x1 + off; bo[1] = y1 + off; bo[2] = x2 + off; bo[3] = y2 + off;
}

// ---- pass 4: TDM stage -> WMMA-tiled IoU sup matrix -> register serial NMS
__global__ void __launch_bounds__(1024)
k_nms(const float* __restrict__ boffw, const float* __restrict__ topScore,
      unsigned* __restrict__ sup, float* __restrict__ out_keep) {
    __shared__ float4 lds_box[KPAD];    // 16 KB, staged by TDM
    __shared__ float  lds_area[KPAD];   // 4 KB
    const int b = blockIdx.x;
    const int tid = threadIdx.x;
    const int lane = tid & 31;
    const int wave = tid >> 5;
    unsigned* __restrict__ srow = sup + (size_t)b * KPAD * 32;

    // --- Tensor Data Mover: DMA the 1024x4 f32 tile global->LDS (wave 0 only)
    if (tid < 32) {
        unsigned lds_addr = (unsigned)(size_t)(&lds_box[0]);
        unsigned long long ga =
            (unsigned long long)(size_t)(boffw + (size_t)b * KPAD * 4);
        v4u g0 = {0u, 0u, 0u, 0u};
        g0[0] = 1u;                                   // count=1, user descriptor
        g0[1] = lds_addr;                             // lds_addr [63:32]
        g0[2] = (unsigned)(ga & 0xFFFFFFFFull);       // global_addr lo
        g0[3] = (unsigned)((ga >> 32) & 0x1FFFFFFull) // global_addr hi [120:96]
              | 0x80000000u;                          // type=2 ("image") [127:126]
        v8i g1 = {0, 0, 0, 0, 0, 0, 0, 0};
        g1[0] = (2 << 16);                    // data_size=4B, wg_mask=0
        g1[1] = (int)((4096u & 0xFFFFu) << 16);        // tensor_dim0 lo
        g1[2] = (int)((4096u >> 16) & 0xFFFFu) | (1 << 16); // dim0 hi | tensor_dim1=1
        g1[3] = (int)(4096u << 16);           // tile_dim0 = 4096 elements
        g1[4] = 1;                            // tile_dim1 = 1
        g1[5] = 4096;                         // tensor_dim0_stride
        v4i gz4 = {0, 0, 0, 0};
        v8i gz8 = {0, 0, 0, 0, 0, 0, 0, 0};
        __builtin_amdgcn_tensor_load_to_lds(g0, g1, gz4, gz4, gz8, 0);
        __builtin_amdgcn_s_wait_tensorcnt(0);
    }
    __syncthreads();

    for (int r = tid; r < KPAD; r += blockDim.x) {
        float4 bx = lds_box[r];
        lds_area[r] = (bx.z - bx.x) * (bx.w - bx.y);
    }
    __syncthreads();

    // --- suppression matrix: one 16x16 tile per wave-iteration.
    // Denominator tile  D[m][n] = area_i + area_j + 1e-9  via f32 WMMA:
    //   A row m = (area[ti*16+m], 1, 0, 0)   (16x4, lanes<16: K0,K1; lanes>=16: K2,K3)
    //   B col n = (1, area[tj*16+n], 0, 0)   (4x16, V0 lanes<16: K0 row; lanes>=16: K1 row)
    //   C      = 1e-9 broadcast
    const int NT = (TOPK + 15) / 16;          // 63
    for (int t = wave; t < NT * NT; t += 32) {
        int ti = t / NT, tj = t % NT;
        if (tj < ti) continue;                // only j>i bits are ever needed
        v2f afrag;
        afrag[0] = (lane < 16) ? lds_area[ti * 16 + (lane & 15)] : 0.f;
        afrag[1] = (lane < 16) ? 1.f : 0.f;
        v2f bfrag;
        bfrag[0] = (lane < 16) ? 1.f : lds_area[tj * 16 + (lane & 15)];
        bfrag[1] = 0.f;
        v8f cfrag;
#pragma unroll
        for (int v = 0; v < 8; ++v) cfrag[v] = 1e-9f;
        v8f dden = __builtin_amdgcn_wmma_f32_16x16x4_f32(
            false, afrag, false, bfrag, (short)0, cfrag, false, false);

        const int m_base = (lane >> 4) << 3;  // C/D layout: lanes>=16 hold M=8..15
        const int n = lane & 15;
#pragma unroll
        for (int v = 0; v < 8; ++v) {
            int gi = ti * 16 + m_base + v;
            int gj = tj * 16 + n;
            if (gi < TOPK && gj < TOPK && gj > gi) {
                float4 bi = lds_box[gi], bj = lds_box[gj];
                float iw = fminf(bi.z, bj.z) - fmaxf(bi.x, bj.x);
                float ih = fminf(bi.w, bj.w) - fmaxf(bi.y, bj.y);
                float inter = fmaxf(iw, 0.f) * fmaxf(ih, 0.f);
                float iou = inter / (dden[v] - inter);  // area_i+area_j-inter+eps
                if (iou > F_THR_NMS)
                    atomicOr(&srow[(size_t)gi * 32 + (gj >> 5)], 1u << (gj & 31));
            }
        }
    }
    __threadfence();
    __syncthreads();

    // --- serial scan, keep mask register-resident in wave 0 (lane w owns word w)
    if (tid < 32) {
        unsigned keep_w = 0u;
#pragma unroll
        for (int bit = 0; bit < 32; ++bit) {
            int r = lane * 32 + bit;
            float s = (r < TOPK) ? topScore[b * KPAD + r] : 0.f;
            if (s >= F_THR_CONF) keep_w |= (1u << bit);
        }
        unsigned nxt = srow[lane];            // prefetch row 0
        for (int i = 0; i < TOPK; ++i) {
            unsigned row = nxt;
            nxt = (i + 1 < TOPK) ? srow[(size_t)(i + 1) * 32 + lane] : 0u;
            unsigned w = (unsigned)__shfl((int)keep_w, i >> 5);
            if ((w >> (i & 31)) & 1u) keep_w &= ~row;
        }
#pragma unroll
        for (int bit = 0; bit < 32; ++bit) {
            int r = lane * 32 + bit;
            if (r < TOPK)
                out_keep[b * TOPK + r] = ((keep_w >> bit) & 1u) ? 1.f : 0.f;
        }
    }
}

// ---------------------------------------------------------------------------
extern "C" void kernel_launch(void* const* d_in, const int* in_sizes, int n_in,
                              void* d_out, int out_size, void* d_ws, size_t ws_size,
                              hipStream_t stream) {
    (void)in_sizes; (void)n_in; (void)out_size; (void)ws_size;
    const float* p_loc    = (const float*)d_in[0];
    const float* p_conf   = (const float*)d_in[1];
    const float* p_landms = (const float*)d_in[2];
    const float* anchors  = (const float*)d_in[3];
    float* out = (float*)d_out;

    // workspace: msc | topIdx | topScore | boff | sup-bitmatrix   (~6.7 MB)
    float* msc      = (float*)d_ws;
    int*   topIdx   = (int*)(msc + (size_t)B_IMG * N_ANCH);
    float* topScore = (float*)(topIdx + (size_t)B_IMG * KPAD);
    float* boffw    = topScore + (size_t)B_IMG * KPAD;
    unsigned* sup   = (unsigned*)(boffw + (size_t)B_IMG * KPAD * 4);

    const int nsup = B_IMG * KPAD * 32;
    k_zero<<<(nsup + 255) / 256, 256, 0, stream>>>(sup, nsup);

    const int ntot = B_IMG * N_ANCH;
    k_scores<<<(ntot + 255) / 256, 256, 0, stream>>>(p_conf, msc);

    k_topk<<<B_IMG, 1024, 0, stream>>>(msc, topIdx, topScore);
    k_gather<<<B_IMG, 1024, 0, stream>>>(p_loc, p_conf, p_landms, anchors,
                                         topIdx, topScore, out, boffw);
    k_nms<<<B_IMG, 1024, 0, stream>>>(boffw, topScore, sup, out /*keep @ 0*/);
}